// SparseGroup_84756884619598
// MI455X (gfx1250) — compile-verified
//
#include <hip/hip_runtime.h>

// Problem constants (match reference)
#define BB   8
#define NN   16384
#define GG   512
#define KNN  32
#define CC   256
#define CH   1024   // point chunk per KNN pass

// d_out layout: neighborhood | center | flat_idx | patch_feats (flat, float32)
#define OFF_N 0
#define OFF_C (BB*GG*KNN*3)                 // 393216
#define OFF_F (OFF_C + BB*GG*3)             // 405504
#define OFF_P (OFF_F + BB*GG*KNN)           // 536576

typedef __attribute__((ext_vector_type(2))) float v2f;
typedef __attribute__((ext_vector_type(8))) float v8f;
typedef __attribute__((ext_vector_type(4))) int   v4i;

// ---- gfx1250 async global->LDS path (guarded; falls back to float4 copies) ----
#if defined(__has_builtin)
#  if __has_builtin(__builtin_amdgcn_global_load_async_to_lds_b128)
#    define HAVE_ASYNC_LDS 1
#  endif
#endif
#ifndef HAVE_ASYNC_LDS
#  define HAVE_ASYNC_LDS 0
#endif

#if HAVE_ASYNC_LDS
#define GAS __attribute__((address_space(1)))
#define LAS __attribute__((address_space(3)))
__device__ __forceinline__ void async_copy16(const float* gsrc, float* ldst) {
    __builtin_amdgcn_global_load_async_to_lds_b128((GAS v4i*)gsrc, (LAS v4i*)ldst, 0, 0);
}
__device__ __forceinline__ void async_wait0() {
#if __has_builtin(__builtin_amdgcn_s_wait_asynccnt)
    __builtin_amdgcn_s_wait_asynccnt(0);
#else
    asm volatile("s_wait_asynccnt 0x0" ::: "memory");
#endif
}
#endif

// ---------------------------------------------------------------------------
// Kernel 1: Furthest Point Sampling. One workgroup per batch.
// Whole point cloud cached interleaved in LDS (192KB of the 320KB WGP LDS),
// filled via coalesced async b128 global->LDS copies. min_d in registers.
// Argmax tie-break: larger value wins; tie -> smaller index (jnp.argmax).
// ---------------------------------------------------------------------------
__global__ __launch_bounds__(1024) void fps_kernel(const float* __restrict__ xyz,
                                                   float* __restrict__ out) {
    __shared__ alignas(16) float spts[NN * 3];   // [N][3] interleaved
    __shared__ float wv[32];
    __shared__ int   wi[32];
    __shared__ int   sFar;

    const int b = blockIdx.x;
    const int t = threadIdx.x;
    const float* X = xyz + (size_t)b * NN * 3;

#if HAVE_ASYNC_LDS
    for (int v = t; v < NN * 3 / 4; v += 1024) async_copy16(X + v * 4, &spts[v * 4]);
    if (t == 0) sFar = 0;
    async_wait0();
    __syncthreads();
#else
    {
        const float4* Xv = (const float4*)X;
        for (int v = t; v < NN * 3 / 4; v += 1024) ((float4*)spts)[v] = Xv[v];
    }
    if (t == 0) sFar = 0;
    __syncthreads();
#endif

    float md[16];
#pragma unroll
    for (int i = 0; i < 16; ++i) md[i] = 3.402823466e38f;

    float* centerOut = out + OFF_C + (size_t)b * GG * 3;
    const int wid  = t >> 5;
    const int lane = t & 31;

    for (int it = 0; it < GG; ++it) {
        const int far = sFar;
        const float cx = spts[3 * far + 0];
        const float cy = spts[3 * far + 1];
        const float cz = spts[3 * far + 2];
        if (t == 0) {
            centerOut[it * 3 + 0] = cx;
            centerOut[it * 3 + 1] = cy;
            centerOut[it * 3 + 2] = cz;
        }
        float bestv = -1.0f;
        int   besti = 0;
#pragma unroll
        for (int i = 0; i < 16; ++i) {
            const int j = t + i * 1024;                 // stride-3 dword reads: conflict-free
            const float dx = spts[3 * j + 0] - cx;
            const float dy = spts[3 * j + 1] - cy;
            const float dz = spts[3 * j + 2] - cz;
            const float d  = dx * dx + dy * dy + dz * dz;
            const float m  = fminf(md[i], d);
            md[i] = m;
            if (m > bestv) { bestv = m; besti = j; }    // strict > keeps earliest j
        }
        // wave32 argmax reduce
        for (int off = 16; off > 0; off >>= 1) {
            const float ov = __shfl_xor(bestv, off, 32);
            const int   oi = __shfl_xor(besti, off, 32);
            if (ov > bestv || (ov == bestv && oi < besti)) { bestv = ov; besti = oi; }
        }
        if (lane == 0) { wv[wid] = bestv; wi[wid] = besti; }
        __syncthreads();
        if (wid == 0) {
            float v  = wv[lane];
            int   ix = wi[lane];
            for (int off = 16; off > 0; off >>= 1) {
                const float ov = __shfl_xor(v, off, 32);
                const int   oi = __shfl_xor(ix, off, 32);
                if (ov > v || (ov == v && oi < ix)) { v = ov; ix = oi; }
            }
            if (lane == 0) sFar = ix;
        }
        __syncthreads();
    }
}

// ---------------------------------------------------------------------------
// Kernel 2: KNN via V_WMMA_F32_16X16X4_F32 + wave-sorted top-32.
// One workgroup (16 waves / 512 threads) per (batch, 16-center tile).
//   A row m (K=0..3) = (cx, cy, cz, |c|^2)
//   B col n (K=0..3) = (-2px, -2py, -2pz, 1)
//   C init           = |p|^2 (per column == per lane)
//   => D = |c|^2 + |p|^2 - 2 c.p  (full 16x16 d2 tile per WMMA)
// Point chunks are double-buffered in LDS via async b128 loads: chunk ch+1
// streams in (ASYNCcnt) while WMMA consumes chunk ch. d2 tiles staged in
// LDS [16][CH+4]; wave w merges candidates of center w into a lane-
// distributed sorted top-32 (lane 0 = nearest), lexicographic (d2, idx) to
// reproduce lax.top_k ordering + tie-breaking.
// ---------------------------------------------------------------------------
__global__ __launch_bounds__(512) void knn_kernel(const float* __restrict__ xyz,
                                                  float* __restrict__ out) {
    __shared__ float cA[16][4];
    __shared__ float d2buf[16][CH + 4];              // +4 pad: de-conflict hi/lo halves
    __shared__ alignas(16) float pts[2][CH * 3];     // double-buffered point chunks

    const int b  = blockIdx.x >> 5;                  // 32 center-tiles per batch
    const int gt = blockIdx.x & 31;
    const int g0 = gt * 16;
    const int t    = threadIdx.x;
    const int w    = t >> 5;
    const int lane = t & 31;

    const float* X   = xyz + (size_t)b * NN * 3;
    const float* CTR = out + OFF_C + ((size_t)b * GG + g0) * 3;

    // Kick chunk 0 into buffer 0
#if HAVE_ASYNC_LDS
    for (int v = t; v < CH * 3 / 4; v += 512) async_copy16(X + v * 4, &pts[0][v * 4]);
#else
    for (int v = t; v < CH * 3 / 4; v += 512) ((float4*)pts[0])[v] = ((const float4*)X)[v];
#endif

    if (t < 16) {
        const float cx = CTR[t * 3 + 0];
        const float cy = CTR[t * 3 + 1];
        const float cz = CTR[t * 3 + 2];
        cA[t][0] = cx; cA[t][1] = cy; cA[t][2] = cz;
        cA[t][3] = cx * cx + cy * cy + cz * cz;
    }

    const int  m  = lane & 15;
    const bool hi = lane >= 16;

    float listVal = 3.402823466e38f;   // sorted ascending across lanes
    int   listIdx = 0;
    v2f afrag;

    for (int ch = 0; ch < NN / CH; ++ch) {
        const int cur    = ch & 1;
        const int pchunk = ch * CH;
#if HAVE_ASYNC_LDS
        async_wait0();
#endif
        __syncthreads();   // pts[cur] + cA ready; previous phase-2 d2buf reads done

        if (ch == 0) {
            // A-frag (32-bit A 16x4): lanes 0-15 hold K=0,1 ; lanes 16-31 hold K=2,3
            afrag.x = hi ? cA[m][2] : cA[m][0];
            afrag.y = hi ? cA[m][3] : cA[m][1];
        }
        if (ch + 1 < NN / CH) {  // stream next chunk while computing this one
            const float* src = X + (size_t)(ch + 1) * CH * 3;
            float* dstb = pts[cur ^ 1];
#if HAVE_ASYNC_LDS
            for (int v = t; v < CH * 3 / 4; v += 512) async_copy16(src + v * 4, &dstb[v * 4]);
#else
            for (int v = t; v < CH * 3 / 4; v += 512) ((float4*)dstb)[v] = ((const float4*)src)[v];
#endif
        }

        // ---- Phase 1: distances (4 WMMA tiles per wave) ----
#pragma unroll
        for (int s = 0; s < 4; ++s) {
            const int tile = w * 4 + s;
            const int pb   = (tile * 16 + m) * 3;
            const float px = pts[cur][pb + 0];
            const float py = pts[cur][pb + 1];
            const float pz = pts[cur][pb + 2];
            const float nrm = px * px + py * py + pz * pz;
            v2f bfrag;   // B 4x16: lanes 0-15 K=0,1 ; lanes 16-31 K=2,3
            bfrag.x = hi ? (-2.0f * pz) : (-2.0f * px);
            bfrag.y = hi ? 1.0f         : (-2.0f * py);
            v8f acc;
#pragma unroll
            for (int r = 0; r < 8; ++r) acc[r] = nrm;
            acc = __builtin_amdgcn_wmma_f32_16x16x4_f32(
                false, afrag, false, bfrag, (short)0, acc, false, false);
            const int col   = tile * 16 + m;
            const int rbase = hi ? 8 : 0;
#pragma unroll
            for (int r = 0; r < 8; ++r) d2buf[rbase + r][col] = acc[r];
        }
        __syncthreads();

        // ---- Phase 2: wave w merges candidates of center w ----
        float lv31 = __shfl(listVal, 31, 32);
        for (int i = 0; i < CH / 32; ++i) {
            const int   li = i * 32 + lane;
            const float c  = d2buf[w][li];
            const int   gi = pchunk + li;
            unsigned long long mask = __ballot(c < lv31);
            while (mask) {
                const int s2 = __builtin_ctzll(mask);
                mask &= mask - 1;
                const float cv = __shfl(c, s2, 32);
                const int   ci = __shfl(gi, s2, 32);
                if (cv < lv31) {
                    const int pos = __popcll(__ballot(listVal <= cv));
                    const float upV = __shfl_up(listVal, 1, 32);
                    const int   upI = __shfl_up(listIdx, 1, 32);
                    listVal = (lane == pos) ? cv : (lane > pos ? upV : listVal);
                    listIdx = (lane == pos) ? ci : (lane > pos ? upI : listIdx);
                    lv31 = __shfl(listVal, 31, 32);
                }
            }
        }
    }

    // ---- Output: neighborhood coords + flat_idx (as exact float) ----
    const int g   = g0 + w;
    const int row = b * GG + g;
    const int src = listIdx;                 // this lane's k-th neighbor (k = lane)
    const int fi  = b * NN + src;
    float* on = out + OFF_N + ((size_t)row * KNN + lane) * 3;
    on[0] = X[src * 3 + 0];
    on[1] = X[src * 3 + 1];
    on[2] = X[src * 3 + 2];
    out[OFF_F + (size_t)row * KNN + lane] = (float)fi;
}

// ---------------------------------------------------------------------------
// Kernel 3: patch_feats gather. 256-thread block copies 4x 1KB rows
// (float4, fully coalesced). feats (128MB) is L2-resident (192MB L2).
// ---------------------------------------------------------------------------
__global__ __launch_bounds__(256) void gather_kernel(const float* __restrict__ feats,
                                                     float* __restrict__ out) {
    const int r  = blockIdx.x * 4 + (threadIdx.x >> 6);   // 0 .. B*G*K-1
    const int t  = threadIdx.x & 63;
    const int fi = (int)out[OFF_F + (size_t)r];           // exact (< 2^24)
    const float4* src = (const float4*)(feats + (size_t)fi * CC);
    float4*       dst = (float4*)(out + OFF_P + (size_t)r * CC);
    dst[t] = src[t];
}

extern "C" void kernel_launch(void* const* d_in, const int* in_sizes, int n_in,
                              void* d_out, int out_size, void* d_ws, size_t ws_size,
                              hipStream_t stream) {
    const float* xyz   = (const float*)d_in[0];
    const float* feats = (const float*)d_in[1];
    float*       out   = (float*)d_out;

    fps_kernel   <<<dim3(BB),                dim3(1024), 0, stream>>>(xyz, out);
    knn_kernel   <<<dim3(BB * 32),           dim3(512),  0, stream>>>(xyz, out);
    gather_kernel<<<dim3(BB * GG * KNN / 4), dim3(256),  0, stream>>>(feats, out);
}